// CriterionIFV_44006234915495
// MI455X (gfx1250) — compile-verified
//
#include <hip/hip_runtime.h>
#include <math.h>

// Problem constants (from the reference setup)
#define B_    8
#define C_    256
#define H_    64
#define W_    128
#define P_    (H_ * W_)      // 8192 pixels
#define NCLS  19
#define KPAD  32             // classes padded to 2 WMMA M-tiles

typedef float v2f __attribute__((ext_vector_type(2)));
typedef float v8f __attribute__((ext_vector_type(8)));

// ---- workspace layout (bytes) ----
#define ACC_OFF      0                                   // 1 float
#define COUNTS_OFF   256                                 // B*NCLS floats
#define MEANS_S_OFF  4096                                // B*C*KPAD floats, zero padded
#define MEANS_T_OFF  (MEANS_S_OFF + B_*C_*KPAD*4)
#define ZERO_BYTES   (MEANS_T_OFF + B_*C_*KPAD*4)        // memset region [0, ZERO_BYTES)
#define LABELS_OFF   ZERO_BYTES                          // B*P ints
#define SUMS_S_OFF   (LABELS_OFF + B_*P_*4)              // B*C*NCLS floats
#define SUMS_T_OFF   (SUMS_S_OFF + B_*C_*NCLS*4)
#define MNORM_S_OFF  (SUMS_T_OFF + B_*C_*NCLS*4)         // B*NCLS floats
#define MNORM_T_OFF  (MNORM_S_OFF + 1024)

// ---------------------------------------------------------------------------
// Kernel 1: labels = argmax_k target[b,p,k] (first-max wins, like jnp.argmax),
// plus per-class counts (LDS-staged, then global float atomics).
// Grid: B*P/256 blocks of 256; each block lies within one batch.
// ---------------------------------------------------------------------------
__global__ __launch_bounds__(256) void k_labels(const float* __restrict__ target,
                                                int* __restrict__ labels,
                                                float* __restrict__ counts) {
  __shared__ float cnt[NCLS];
  int tid = threadIdx.x;
  if (tid < NCLS) cnt[tid] = 0.0f;
  __syncthreads();
  int idx = blockIdx.x * 256 + tid;                 // [0, B*P)
  const float* t = target + (size_t)idx * NCLS;     // per-pixel contiguous 19 floats
  float best = t[0];
  int arg = 0;
#pragma unroll
  for (int k = 1; k < NCLS; ++k) {
    float v = t[k];
    if (v > best) { best = v; arg = k; }
  }
  labels[idx] = arg;
  atomicAdd(&cnt[arg], 1.0f);
  __syncthreads();
  int b = idx / P_;                                 // uniform per block
  if (tid < NCLS) atomicAdd(&counts[b * NCLS + tid], cnt[tid]);
}

// ---------------------------------------------------------------------------
// Kernel 2: class sums. One wave per (b, channel); lanes stride over pixels,
// 19 register accumulators per feature, shfl tree reduction at the end.
// Labels (32 KB/batch) are L2-resident across the C re-reads.
// Grid: B*(C/8) blocks of 256 (8 waves).
// ---------------------------------------------------------------------------
__global__ __launch_bounds__(256) void k_sums(const float* __restrict__ fS,
                                              const float* __restrict__ fT,
                                              const int* __restrict__ labels,
                                              float* __restrict__ sumsS,
                                              float* __restrict__ sumsT) {
  int wave = threadIdx.x >> 5, lane = threadIdx.x & 31;
  int b = blockIdx.x / (C_ / 8);
  int c = (blockIdx.x % (C_ / 8)) * 8 + wave;
  const float* rs = fS + ((size_t)b * C_ + c) * P_;
  const float* rt = fT + ((size_t)b * C_ + c) * P_;
  const int*   lb = labels + (size_t)b * P_;
  float aS[NCLS], aT[NCLS];
#pragma unroll
  for (int k = 0; k < NCLS; ++k) { aS[k] = 0.0f; aT[k] = 0.0f; }
  for (int p = lane; p < P_; p += 32) {
    float vS = rs[p], vT = rt[p];
    int l = lb[p];
#pragma unroll
    for (int k = 0; k < NCLS; ++k) {
      bool m = (l == k);
      aS[k] += m ? vS : 0.0f;
      aT[k] += m ? vT : 0.0f;
    }
  }
#pragma unroll
  for (int k = 0; k < NCLS; ++k) {
    float s = aS[k], t2 = aT[k];
#pragma unroll
    for (int off = 16; off; off >>= 1) {
      s  += __shfl_down(s,  off, 32);
      t2 += __shfl_down(t2, off, 32);
    }
    if (lane == 0) {
      size_t o = ((size_t)b * C_ + c) * NCLS + k;
      sumsS[o] = s;
      sumsT[o] = t2;
    }
  }
}

// ---------------------------------------------------------------------------
// Kernel 3: means (written K-padded to 32, pad pre-zeroed by memset) and
// clamped center norms  mn[b,k] = max(||means[b,:,k]||, 1e-8).
// Grid: B*NCLS blocks of 256 (one thread per channel).
// ---------------------------------------------------------------------------
__global__ __launch_bounds__(256) void k_means(const float* __restrict__ sumsS,
                                               const float* __restrict__ sumsT,
                                               const float* __restrict__ counts,
                                               float* __restrict__ meansPS,
                                               float* __restrict__ meansPT,
                                               float* __restrict__ mnS,
                                               float* __restrict__ mnT) {
  int b = blockIdx.x / NCLS, k = blockIdx.x % NCLS, c = threadIdx.x;
  float inv = 1.0f / (counts[b * NCLS + k] + 1e-6f);
  size_t si = ((size_t)b * C_ + c) * NCLS + k;
  size_t mi = ((size_t)b * C_ + c) * KPAD + k;
  float mS = sumsS[si] * inv, mT = sumsT[si] * inv;
  meansPS[mi] = mS;
  meansPT[mi] = mT;
  __shared__ float redS[256], redT[256];
  redS[c] = mS * mS; redT[c] = mT * mT;
  __syncthreads();
  for (int s = 128; s; s >>= 1) {
    if (c < s) { redS[c] += redS[c + s]; redT[c] += redT[c + s]; }
    __syncthreads();
  }
  if (c == 0) {
    mnS[b * NCLS + k] = fmaxf(sqrtf(redS[0]), 1e-8f);
    mnT[b * NCLS + k] = fmaxf(sqrtf(redT[0]), 1e-8f);
  }
}

// ---------------------------------------------------------------------------
// Kernel 4 (WMMA): G[k,p] = means^T (KPAD x C) * f (C x P) via
// V_WMMA_F32_16X16X4_F32, exact fp32 math. Each wave owns one 16-pixel tile
// and two 16-class M-tiles, accumulating over C=256 in 64 steps of 4.
//   A frag (means^T): lane l<16 -> {means[c0+0][l], means[c0+1][l]},
//                     lane l>=16 -> channels c0+2,c0+3; served from LDS in a
//                     swizzled layout [g][class][4ch] -> conflict-free b64.
//   B frag (f):       lane l<16 -> f[c0][p0+l] / f[c0+1][p0+l]; lanes>=16
//                     rows c0+2/c0+3 -> two coalesced 64B segments per load.
// ||f_p||^2 accumulated from the same B registers; halves combined with
// shfl_xor(16). Tiles spilled to per-wave LDS, dot gathered at class=lab.
// Grid: B*P/128 blocks of 256 (8 waves x 16 pixels).
// ---------------------------------------------------------------------------
__global__ __launch_bounds__(256) void k_cosloss(const float* __restrict__ fS,
                                                 const float* __restrict__ fT,
                                                 const float* __restrict__ meansPS,
                                                 const float* __restrict__ meansPT,
                                                 const float* __restrict__ mnS,
                                                 const float* __restrict__ mnT,
                                                 const int* __restrict__ labels,
                                                 float* __restrict__ acc) {
  __shared__ float sMean[C_ / 4 * KPAD * 4 / 4];   // 2048 floats (8 KB)
  __shared__ float sG[8 * KPAD * 16];              // 8 waves * 32x16 tile (16 KB)

  int tid  = threadIdx.x;
  int b    = blockIdx.x >> 6;                      // 64 blocks per batch
  int pblk = (blockIdx.x & 63) * 128;
  int wave = tid >> 5, lane = tid & 31;
  int px = lane & 15, hi = lane >> 4;              // pixel-in-tile, half-wave
  int p  = pblk + wave * 16 + px;

  const float* feats[2] = {fS, fT};
  const float* mpads[2] = {meansPS, meansPT};
  const float* mns[2]   = {mnS, mnT};
  float* gbase = &sG[wave * (KPAD * 16)];
  int lab = labels[b * P_ + p];
  float cosv[2];

  for (int f = 0; f < 2; ++f) {
    // stage this feature's per-batch means into swizzled LDS:
    // sMean[g*128 + k*4 + j] = means[b, c=4g+j, k]
    const float* mp = mpads[f];
    for (int i = tid; i < 8192; i += 256) {
      int g = i >> 7, r = i & 127, k = r >> 2, j = r & 3;
      sMean[i] = mp[((size_t)b * C_ + (g << 2) + j) * KPAD + k];
    }
    __syncthreads();

    const float* fp = feats[f] + (size_t)b * C_ * P_ + p;
    v8f acc0 = {}; v8f acc1 = {};
    float nrm = 0.0f;
    for (int g = 0; g < 64; ++g) {                 // c0 = 4*g
      int crow = (g << 2) + (hi << 1);
      v2f bv;
      bv.x = fp[(size_t)crow * P_];
      bv.y = fp[(size_t)(crow + 1) * P_];
      nrm += bv.x * bv.x + bv.y * bv.y;
      int abase = (g << 7) + (hi << 1);
      v2f a0, a1;
      a0.x = sMean[abase + px * 4];
      a0.y = sMean[abase + px * 4 + 1];
      a1.x = sMean[abase + (px + 16) * 4];
      a1.y = sMean[abase + (px + 16) * 4 + 1];
      acc0 = __builtin_amdgcn_wmma_f32_16x16x4_f32(false, a0, false, bv,
                                                   (short)0, acc0, false, false);
      acc1 = __builtin_amdgcn_wmma_f32_16x16x4_f32(false, a1, false, bv,
                                                   (short)0, acc1, false, false);
    }
    // spill D tiles: VGPR r -> (class r + 8*hi, pixel px) / tile1 classes +16
#pragma unroll
    for (int r = 0; r < 8; ++r) {
      int cls = r + (hi << 3);
      gbase[cls * 16 + px]        = acc0[r];
      gbase[(cls + 16) * 16 + px] = acc1[r];
    }
    float nfull = nrm + __shfl_xor(nrm, 16, 32);   // combine channel halves
    float dot = gbase[lab * 16 + px];
    float nf  = fmaxf(sqrtf(nfull), 1e-8f);
    cosv[f] = dot / (nf * mns[f][b * NCLS + lab]);
    __syncthreads();                               // sMean reused next feature
  }

  float d = cosv[0] - cosv[1];
  float val = (hi == 0) ? d * d : 0.0f;            // lanes>=16 are duplicates
#pragma unroll
  for (int off = 16; off; off >>= 1) val += __shfl_xor(val, off, 32);
  if (lane == 0) atomicAdd(acc, val);
}

// ---------------------------------------------------------------------------
// Kernel 5: finalize scalar mean.
// ---------------------------------------------------------------------------
__global__ void k_final(const float* __restrict__ acc, float* __restrict__ out) {
  out[0] = acc[0] * (1.0f / (float)(B_ * P_));
}

extern "C" void kernel_launch(void* const* d_in, const int* in_sizes, int n_in,
                              void* d_out, int out_size, void* d_ws, size_t ws_size,
                              hipStream_t stream) {
  (void)in_sizes; (void)n_in; (void)out_size; (void)ws_size;
  const float* feat_S = (const float*)d_in[0];
  const float* feat_T = (const float*)d_in[1];
  const float* target = (const float*)d_in[2];
  float* out = (float*)d_out;
  char* ws = (char*)d_ws;

  float* acc     = (float*)(ws + ACC_OFF);
  float* counts  = (float*)(ws + COUNTS_OFF);
  float* meansPS = (float*)(ws + MEANS_S_OFF);
  float* meansPT = (float*)(ws + MEANS_T_OFF);
  int*   labels  = (int*)  (ws + LABELS_OFF);
  float* sumsS   = (float*)(ws + SUMS_S_OFF);
  float* sumsT   = (float*)(ws + SUMS_T_OFF);
  float* mnS     = (float*)(ws + MNORM_S_OFF);
  float* mnT     = (float*)(ws + MNORM_T_OFF);

  // zero accumulator, counts, and padded means (pad lanes k>=19 must be 0)
  hipMemsetAsync(ws, 0, (size_t)ZERO_BYTES, stream);

  k_labels <<<(B_ * P_) / 256, 256, 0, stream>>>(target, labels, counts);
  k_sums   <<<B_ * (C_ / 8),   256, 0, stream>>>(feat_S, feat_T, labels, sumsS, sumsT);
  k_means  <<<B_ * NCLS,       256, 0, stream>>>(sumsS, sumsT, counts,
                                                 meansPS, meansPT, mnS, mnT);
  k_cosloss<<<(B_ * P_) / 128, 256, 0, stream>>>(feat_S, feat_T, meansPS, meansPT,
                                                 mnS, mnT, labels, acc);
  k_final  <<<1, 1, 0, stream>>>(acc, out);
}